// SimpleGCNLayer_43164421325634
// MI455X (gfx1250) — compile-verified
//
#include <hip/hip_runtime.h>

typedef float v2f __attribute__((ext_vector_type(2)));
typedef float v4f __attribute__((ext_vector_type(4)));
typedef float v8f __attribute__((ext_vector_type(8)));

#define BB 8
#define NN 256
#define DD 128
#define ROWS (BB * NN * NN)   // 524288 edge-feature rows of length 128
#define M2 (BB * NN)          // 2048 node rows

// ---------------------------------------------------------------------------
// Kernel 1: fused  w_raw[b,i,j] = ||e[b,i,j,:]||_2   +   copy e -> out_e
// One wave32 per 128-float row: lane loads float4 (16B), wave-reduce sum-sq.
// e (256MB) > L2 (192MB) and is touched exactly once: non-temporal on BOTH
// the load and the store so the 512MB stream does not evict the small
// reusable working set (h, W, wP, X).
// ---------------------------------------------------------------------------
__global__ __launch_bounds__(256) void norm_copy(const float* __restrict__ e,
                                                 float* __restrict__ e_out,
                                                 float* __restrict__ w_raw) {
  int gw   = (blockIdx.x * blockDim.x + threadIdx.x) >> 5;  // global wave = row
  int lane = threadIdx.x & 31;
  const v4f* src = (const v4f*)(e + (size_t)gw * DD) + lane;
  v4f v = __builtin_nontemporal_load(src);
  __builtin_nontemporal_store(v, (v4f*)(e_out + (size_t)gw * DD) + lane);
  float s = v.x * v.x + v.y * v.y + v.z * v.z + v.w * v.w;
  s += __shfl_xor(s, 16, 32);
  s += __shfl_xor(s, 8, 32);
  s += __shfl_xor(s, 4, 32);
  s += __shfl_xor(s, 2, 32);
  s += __shfl_xor(s, 1, 32);
  if (lane == 0) w_raw[gw] = sqrtf(s);
}

// ---------------------------------------------------------------------------
// Kernel 2: softmax over j (row length 256), in-place. One block per (b,i).
// ---------------------------------------------------------------------------
__global__ __launch_bounds__(256) void softmax256(float* __restrict__ w) {
  __shared__ float red[256];
  int tid = threadIdx.x;
  size_t base = (size_t)blockIdx.x * NN;
  float x = w[base + tid];
  red[tid] = x;
  __syncthreads();
  for (int s = 128; s > 0; s >>= 1) {
    if (tid < s) red[tid] = fmaxf(red[tid], red[tid + s]);
    __syncthreads();
  }
  float mx = red[0];
  __syncthreads();
  float ex = __expf(x - mx);
  red[tid] = ex;
  __syncthreads();
  for (int s = 128; s > 0; s >>= 1) {
    if (tid < s) red[tid] += red[tid + s];
    __syncthreads();
  }
  w[base + tid] = ex / red[0];
}

// ---------------------------------------------------------------------------
// Kernel 3: agg = P @ H per batch (256x256 * 256x128), fused X = h + agg.
// Full-fp32 WMMA 16x16x4, software-pipelined fragment loads (double buffer)
// so the backend can overlap v_wmma with the next K-step's global loads
// instead of a full s_wait_loadcnt 0 every step.
// A frag (16x4 f32): lane = M + 16*(K/2), vgpr = K%2  -> contiguous float2.
// B frag (4x16 f32): lane = N + 16*(K/2), vgpr = K%2  -> column loads.
// C/D (16x16 f32):   vgpr r: lanes 0-15 -> (M=r, N=lane); 16-31 -> (M=r+8).
// ---------------------------------------------------------------------------
__global__ __launch_bounds__(128) void gemm1_agg(const float* __restrict__ wP,
                                                 const float* __restrict__ h,
                                                 float* __restrict__ X) {
  int wave = (blockIdx.x * blockDim.x + threadIdx.x) >> 5;
  int lane = threadIdx.x & 31;
  int hs = lane >> 4;       // half-wave select (K pair / M+8 select)
  int lm = lane & 15;
  int b   = wave >> 7;             // 128 tiles per batch
  int rem = wave & 127;
  int i0  = (rem >> 3) << 4;       // 16 row tiles
  int d0  = (rem & 7) << 4;        // 8 col tiles
  const float* P = wP + (size_t)b * NN * NN + (size_t)(i0 + lm) * NN + 2 * hs;
  const float* H = h + (size_t)b * NN * DD;
  const float* Hc = H + (size_t)(2 * hs) * DD + d0 + lm;  // column base
  v8f c = {};

  v2f a0 = *(const v2f*)(P);
  v2f b0;
  b0.x = Hc[0];
  b0.y = Hc[DD];
  for (int k0 = 0; k0 < NN - 4; k0 += 4) {
    v2f a1 = *(const v2f*)(P + k0 + 4);
    v2f b1;
    b1.x = Hc[(size_t)(k0 + 4) * DD];
    b1.y = Hc[(size_t)(k0 + 5) * DD];
    c = __builtin_amdgcn_wmma_f32_16x16x4_f32(false, a0, false, b0,
                                              (short)0, c, false, false);
    a0 = a1;
    b0 = b1;
  }
  c = __builtin_amdgcn_wmma_f32_16x16x4_f32(false, a0, false, b0,
                                            (short)0, c, false, false);

  float* Xb = X + (size_t)b * NN * DD;
  for (int r = 0; r < 8; ++r) {
    int row = i0 + r + 8 * hs;
    int col = d0 + lm;
    Xb[(size_t)row * DD + col] = c[r] + H[(size_t)row * DD + col];
  }
}

// ---------------------------------------------------------------------------
// Kernel 4: Y = relu(X @ W^T + bias), Y -> d_out h-region; fused BN partial
// sums (sum, sumsq per feature) via global f32 atomics (128 counters).
// Y[m,k] = sum_d X[m,d]*W[k,d]; both fragments are contiguous float2 loads.
// ---------------------------------------------------------------------------
__global__ __launch_bounds__(128) void gemm2_linrelu(const float* __restrict__ X,
                                                     const float* __restrict__ W,
                                                     const float* __restrict__ bias,
                                                     float* __restrict__ Y,
                                                     float* __restrict__ sums,
                                                     float* __restrict__ sumsq) {
  int wave = (blockIdx.x * blockDim.x + threadIdx.x) >> 5;
  int lane = threadIdx.x & 31;
  int hs = lane >> 4;
  int lm = lane & 15;
  int m0 = (wave >> 3) << 4;   // 128 row tiles over B*N = 2048
  int k0 = (wave & 7) << 4;    // 8 col tiles over 128 features
  const float* Xr = X + (size_t)(m0 + lm) * DD + 2 * hs;
  const float* Wr = W + (size_t)(k0 + lm) * DD + 2 * hs;
  v8f c = {};

  v2f a0 = *(const v2f*)(Xr);
  v2f b0 = *(const v2f*)(Wr);
  for (int kk = 0; kk < DD - 4; kk += 4) {
    v2f a1 = *(const v2f*)(Xr + kk + 4);
    v2f b1 = *(const v2f*)(Wr + kk + 4);
    c = __builtin_amdgcn_wmma_f32_16x16x4_f32(false, a0, false, b0,
                                              (short)0, c, false, false);
    a0 = a1;
    b0 = b1;
  }
  c = __builtin_amdgcn_wmma_f32_16x16x4_f32(false, a0, false, b0,
                                            (short)0, c, false, false);

  float bv = bias[k0 + lm];
  float ls = 0.f, lq = 0.f;
  for (int r = 0; r < 8; ++r) {
    int row = m0 + r + 8 * hs;
    float y = fmaxf(c[r] + bv, 0.f);
    Y[(size_t)row * DD + k0 + lm] = y;
    ls += y;
    lq += y * y;
  }
  atomicAdd(&sums[k0 + lm], ls);
  atomicAdd(&sumsq[k0 + lm], lq);
}

// Zero the 256 contiguous stat floats (sums + sumsq) each call.
__global__ void zero256(float* __restrict__ p) { p[threadIdx.x] = 0.f; }

// Fold BN stats + affine into per-feature scale/shift.
__global__ void bn_stats(const float* __restrict__ sums,
                         const float* __restrict__ sumsq,
                         const float* __restrict__ gamma,
                         const float* __restrict__ beta,
                         float* __restrict__ scale,
                         float* __restrict__ shift) {
  int k = threadIdx.x;
  const float inv = 1.0f / (float)M2;
  float mean = sums[k] * inv;
  float var  = sumsq[k] * inv - mean * mean;
  float sc   = rsqrtf(var + 1e-5f) * gamma[k];
  scale[k] = sc;
  shift[k] = beta[k] - mean * sc;
}

// In-place: out = y*scale + shift + h
__global__ __launch_bounds__(256) void bn_apply(float* __restrict__ Y,
                                                const float* __restrict__ h,
                                                const float* __restrict__ scale,
                                                const float* __restrict__ shift) {
  int idx = blockIdx.x * blockDim.x + threadIdx.x;
  int k = idx & (DD - 1);
  float y = Y[idx];
  Y[idx] = y * scale[k] + shift[k] + h[idx];
}

extern "C" void kernel_launch(void* const* d_in, const int* in_sizes, int n_in,
                              void* d_out, int out_size, void* d_ws, size_t ws_size,
                              hipStream_t stream) {
  const float* h     = (const float*)d_in[0];
  const float* e     = (const float*)d_in[1];
  const float* W     = (const float*)d_in[2];
  const float* bias  = (const float*)d_in[3];
  const float* gamma = (const float*)d_in[4];
  const float* beta  = (const float*)d_in[5];

  float* out_h = (float*)d_out;                       // (B,N,D) h_new
  float* out_e = out_h + (size_t)BB * NN * DD;        // copy of e

  float* ws    = (float*)d_ws;
  float* wP    = ws;                                  // B*N*N softmax probs (2MB)
  float* X     = wP + (size_t)ROWS;                   // B*N*D  (1MB)
  float* sums  = X + (size_t)BB * NN * DD;            // 128
  float* sumsq = sums + DD;                           // 128 (contiguous with sums)
  float* scale = sumsq + DD;                          // 128
  float* shift = scale + DD;                          // 128

  norm_copy<<<ROWS / 8, 256, 0, stream>>>(e, out_e, wP);
  softmax256<<<BB * NN, 256, 0, stream>>>(wP);
  gemm1_agg<<<(BB * (NN / 16) * (DD / 16)) / 4, 128, 0, stream>>>(wP, h, X);
  zero256<<<1, 256, 0, stream>>>(sums);
  gemm2_linrelu<<<((M2 / 16) * (DD / 16)) / 4, 128, 0, stream>>>(X, W, bias, out_h,
                                                                 sums, sumsq);
  bn_stats<<<1, DD, 0, stream>>>(sums, sumsq, gamma, beta, scale, shift);
  bn_apply<<<(BB * NN * DD) / 256, 256, 0, stream>>>(out_h, h, scale, shift);
}